// MPNNModel_62912680952074
// MI455X (gfx1250) — compile-verified
//
#include <hip/hip_runtime.h>

typedef __attribute__((ext_vector_type(2))) float v2f;
typedef __attribute__((ext_vector_type(8))) float v8f;

#define K_B 64
#define K_T 12
#define K_N 512
#define K_D 2
#define K_R 64
#define K_M 64
#define K_H 12
#define K_NB (K_N * K_B)   /* 32768 rows of the (node,batch) "super-batch" */
#define K_BM (K_B * K_M)   /* 4096 columns of the aggregation GEMM */

static __device__ __forceinline__ v8f wmma_f32(v2f a, v2f b, v8f c) {
    // V_WMMA_F32_16X16X4_F32 : D = A(16x4,f32) * B(4x16,f32) + C(16x16,f32)
    return __builtin_amdgcn_wmma_f32_16x16x4_f32(false, a, false, b, (short)0, c,
                                                 false, false);
}

static __device__ __forceinline__ float sigmoidf_(float x) {
    return 1.0f / (1.0f + __expf(-x));
}

// ---------------------------------------------------------------------------
// K0: adjT[i][j] = adj[j][i] > 0 ? adj[j][i] : 0   (once per launch)
// ---------------------------------------------------------------------------
__global__ void k_transpose_adj(const float* __restrict__ adj,
                                float* __restrict__ adjT) {
    int i = blockIdx.x * 16 + threadIdx.x;
    int j = blockIdx.y * 16 + threadIdx.y;
    float v = adj[j * K_N + i];
    adjT[i * K_N + j] = v > 0.0f ? v : 0.0f;
}

// ---------------------------------------------------------------------------
// K1: hw[r, mm] = sum_k h[r,k] * W_msg[mm,k] + b_msg[mm]      r = node*B+b
//     (32768 x 64) = (32768 x 64) * (64 x 64)
// grid.x = NB/16 row-tiles, 4 waves/block cover the 4 column-tiles
// ---------------------------------------------------------------------------
__global__ void k_message(const float* __restrict__ h,
                          const float* __restrict__ Wmsg,
                          const float* __restrict__ bmsg,
                          float* __restrict__ hw) {
    const int lane = threadIdx.x & 31;
    const int wv   = threadIdx.x >> 5;
    const int lm = lane & 15, lh = lane >> 4;
    const int row0 = blockIdx.x * 16;
    const int n0   = wv * 16;

    v8f acc = {};
    const float* arow = h    + (size_t)(row0 + lm) * K_R + 2 * lh;
    const float* brow = Wmsg + (size_t)(n0   + lm) * K_R + 2 * lh; // B[k][n]=Wmsg[n][k]
#pragma unroll
    for (int k0 = 0; k0 < K_R; k0 += 4) {
        v2f a = *(const v2f*)(arow + k0);
        v2f b = *(const v2f*)(brow + k0);
        acc = wmma_f32(a, b, acc);
    }
    const float bias = bmsg[n0 + lm];
#pragma unroll
    for (int v = 0; v < 8; ++v) {
        int row = row0 + v + 8 * lh;              // C layout: VGPR v -> M=v / v+8
        hw[(size_t)row * K_M + n0 + lm] = acc[v] + bias;
    }
}

// ---------------------------------------------------------------------------
// K2: m[i, c] = sum_j adjT[i][j] * hw[j][c]      c = b*M+mm  (512x4096 GEMM)
// grid = (N/16, 4096/64), 4 waves/block, K-loop 512/4 = 128 WMMAs
// ---------------------------------------------------------------------------
__global__ void k_aggregate(const float* __restrict__ adjT,
                            const float* __restrict__ hw,
                            float* __restrict__ mbuf) {
    const int lane = threadIdx.x & 31;
    const int wv   = threadIdx.x >> 5;
    const int lm = lane & 15, lh = lane >> 4;
    const int row0 = blockIdx.x * 16;                 // node-i tile
    const int n0   = (blockIdx.y * 4 + wv) * 16;      // (b,mm) tile

    v8f acc = {};
    const float* arow = adjT + (size_t)(row0 + lm) * K_N + 2 * lh;
    const float* bcol = hw + (size_t)(2 * lh) * K_BM + n0 + lm;
#pragma unroll 4
    for (int k0 = 0; k0 < K_N; k0 += 4) {
        v2f a = *(const v2f*)(arow + k0);
        v2f b;
        b.x = bcol[(size_t)k0 * K_BM];
        b.y = bcol[(size_t)(k0 + 1) * K_BM];
        acc = wmma_f32(a, b, acc);
    }
#pragma unroll
    for (int v = 0; v < 8; ++v) {
        int row = row0 + v + 8 * lh;
        mbuf[(size_t)row * K_BM + n0 + lm] = acc[v];
    }
}

// ---------------------------------------------------------------------------
// K3: fused GRU cell over rows r = node*B+b:
//   gi = [m|x] @ W_ih^T + b_ih   (K = 66)
//   gh =  h    @ W_hh^T + b_hh   (K = 64)
//   r,z,n gates -> h updated in place.
// Each wave: 16 rows x 16 cols-per-gate (6 f32 accumulators).
// ---------------------------------------------------------------------------
__global__ void k_gru(float* __restrict__ h,
                      const float* __restrict__ mbuf,
                      const float* __restrict__ x,
                      const float* __restrict__ Wih,   // [192, 66]
                      const float* __restrict__ Whh,   // [192, 64]
                      const float* __restrict__ bih,
                      const float* __restrict__ bhh,
                      int t) {
    const int lane = threadIdx.x & 31;
    const int wv   = threadIdx.x >> 5;
    const int lm = lane & 15, lh = lane >> 4;
    const int row0 = blockIdx.x * 16;
    const int c0   = wv * 16;                  // column offset inside each gate

    v8f gir = {}, giz = {}, gin = {};
    v8f ghr = {}, ghz = {}, ghn = {};

    const float* ah  = h    + (size_t)(row0 + lm) * K_R + 2 * lh;
    const float* am  = mbuf + (size_t)(row0 + lm) * K_M + 2 * lh;
    const float* bhr = Whh + (size_t)(c0 + lm)           * K_R + 2 * lh;
    const float* bhz = Whh + (size_t)(K_R + c0 + lm)     * K_R + 2 * lh;
    const float* bhn = Whh + (size_t)(2 * K_R + c0 + lm) * K_R + 2 * lh;
    const float* bir = Wih + (size_t)(c0 + lm)           * 66 + 2 * lh;
    const float* biz = Wih + (size_t)(K_R + c0 + lm)     * 66 + 2 * lh;
    const float* bin_= Wih + (size_t)(2 * K_R + c0 + lm) * 66 + 2 * lh;

#pragma unroll 4
    for (int k0 = 0; k0 < K_R; k0 += 4) {            // gh over h (K=64)
        v2f a = *(const v2f*)(ah + k0);
        ghr = wmma_f32(a, *(const v2f*)(bhr + k0), ghr);
        ghz = wmma_f32(a, *(const v2f*)(bhz + k0), ghz);
        ghn = wmma_f32(a, *(const v2f*)(bhn + k0), ghn);
    }
#pragma unroll 4
    for (int k0 = 0; k0 < K_M; k0 += 4) {            // gi over m (K=0..63)
        v2f a = *(const v2f*)(am + k0);
        gir = wmma_f32(a, *(const v2f*)(bir + k0), gir);
        giz = wmma_f32(a, *(const v2f*)(biz + k0), giz);
        gin = wmma_f32(a, *(const v2f*)(bin_ + k0), gin);
    }
    {   // gi tail: K = 64,65 come from x_t; K = 66,67 (lanes 16..31) are zero
        v2f a = {}, br = {}, bz = {}, bn = {};
        if (lh == 0) {
            int row  = row0 + lm;
            int node = row >> 6, bb = row & 63;
            a  = *(const v2f*)(x + (size_t)(((bb * K_T) + t) * K_N + node) * K_D);
            br = *(const v2f*)(Wih + (size_t)(c0 + lm)           * 66 + 64);
            bz = *(const v2f*)(Wih + (size_t)(K_R + c0 + lm)     * 66 + 64);
            bn = *(const v2f*)(Wih + (size_t)(2 * K_R + c0 + lm) * 66 + 64);
        }
        gir = wmma_f32(a, br, gir);
        giz = wmma_f32(a, bz, giz);
        gin = wmma_f32(a, bn, gin);
    }

    const float b_ir = bih[c0 + lm], b_iz = bih[K_R + c0 + lm], b_in = bih[2 * K_R + c0 + lm];
    const float b_hr = bhh[c0 + lm], b_hz = bhh[K_R + c0 + lm], b_hn = bhh[2 * K_R + c0 + lm];

    // All waves in this block read h rows [row0,row0+16) above; they write
    // disjoint 16-column slices of the same rows below -> one barrier.
    __syncthreads();

#pragma unroll
    for (int v = 0; v < 8; ++v) {
        int row = row0 + v + 8 * lh;
        size_t idx = (size_t)row * K_R + c0 + lm;
        float hold = h[idx];
        float r = sigmoidf_(gir[v] + b_ir + ghr[v] + b_hr);
        float z = sigmoidf_(giz[v] + b_iz + ghz[v] + b_hz);
        float n = tanhf(gin[v] + b_in + r * (ghn[v] + b_hn));
        h[idx] = (1.0f - z) * n + z * hold;
    }
}

// ---------------------------------------------------------------------------
// K4: out[b, hh, node, 0] = sum_k h[r,k] * W_out[hh,k] + b_out[hh]
//     (32768 x 12) GEMM, one 16-wide col tile (cols 12..15 padded with 0)
// ---------------------------------------------------------------------------
__global__ void k_output(const float* __restrict__ h,
                         const float* __restrict__ Wout,  // [12, 64]
                         const float* __restrict__ bout,
                         float* __restrict__ out) {
    const int lane = threadIdx.x & 31;
    const int wv   = threadIdx.x >> 5;
    const int lm = lane & 15, lh = lane >> 4;
    const int row0 = (blockIdx.x * 4 + wv) * 16;
    const bool valid = lm < K_H;

    v8f acc = {};
    const float* arow = h + (size_t)(row0 + lm) * K_R + 2 * lh;
    const float* brow = Wout + (size_t)lm * K_R + 2 * lh;
#pragma unroll
    for (int k0 = 0; k0 < K_R; k0 += 4) {
        v2f a = *(const v2f*)(arow + k0);
        v2f b = {};
        if (valid) b = *(const v2f*)(brow + k0);
        acc = wmma_f32(a, b, acc);
    }
    if (valid) {
        const float bias = bout[lm];
#pragma unroll
        for (int v = 0; v < 8; ++v) {
            int row  = row0 + v + 8 * lh;
            int node = row >> 6, bb = row & 63;
            out[(size_t)(bb * K_H + lm) * K_N + node] = acc[v] + bias;
        }
    }
}

// ---------------------------------------------------------------------------
extern "C" void kernel_launch(void* const* d_in, const int* in_sizes, int n_in,
                              void* d_out, int out_size, void* d_ws, size_t ws_size,
                              hipStream_t stream) {
    (void)in_sizes; (void)n_in; (void)out_size; (void)ws_size;

    const float* x    = (const float*)d_in[0];   // [B,T,N,D]
    const float* h0   = (const float*)d_in[1];   // [N,B,R]
    const float* adj  = (const float*)d_in[2];   // [N,N]
    const float* Wmsg = (const float*)d_in[3];   // [M,R]
    const float* bmsg = (const float*)d_in[4];   // [M]
    const float* Wih  = (const float*)d_in[5];   // [3R, D+M]
    const float* Whh  = (const float*)d_in[6];   // [3R, R]
    const float* bih  = (const float*)d_in[7];   // [3R]
    const float* bhh  = (const float*)d_in[8];   // [3R]
    const float* Wout = (const float*)d_in[9];   // [O*H, R]
    const float* bout = (const float*)d_in[10];  // [O*H]
    float* out = (float*)d_out;

    // workspace layout (floats): adjT | h | hw | m  == ~25 MB total
    float* ws   = (float*)d_ws;
    float* adjT = ws;
    float* h    = adjT + (size_t)K_N * K_N;
    float* hw   = h    + (size_t)K_NB * K_R;
    float* mbuf = hw   + (size_t)K_NB * K_M;

    hipMemcpyAsync(h, h0, (size_t)K_NB * K_R * sizeof(float),
                   hipMemcpyDeviceToDevice, stream);
    k_transpose_adj<<<dim3(K_N / 16, K_N / 16), dim3(16, 16), 0, stream>>>(adj, adjT);

    for (int t = 0; t < K_T; ++t) {
        k_message  <<<dim3(K_NB / 16), dim3(128), 0, stream>>>(h, Wmsg, bmsg, hw);
        k_aggregate<<<dim3(K_N / 16, K_BM / 64), dim3(128), 0, stream>>>(adjT, hw, mbuf);
        k_gru      <<<dim3(K_NB / 16), dim3(128), 0, stream>>>(h, mbuf, x, Wih, Whh,
                                                               bih, bhh, t);
    }
    k_output<<<dim3(K_NB / 64), dim3(128), 0, stream>>>(h, Wout, bout, out);
}